// TrackingLoss_31688268710033
// MI455X (gfx1250) — compile-verified
//
#include <hip/hip_runtime.h>
#include <math.h>

// Problem constants (fixed by the reference harness)
#define N_A   65536   // total anchors = 16 * 64 * 64
#define NEG   2048
#define POS   64
#define BS    128

typedef __attribute__((ext_vector_type(2))) float v2f;
typedef __attribute__((ext_vector_type(8))) float v8f;

// Numerically stable softplus: log(1 + exp(d)) == max(d,0) + log1p(exp(-|d|))
__device__ __forceinline__ float softplus_f(float d) {
    return fmaxf(d, 0.0f) + log1pf(expf(-fabsf(d)));
}

// Kernel 1: one workgroup per batch element. 256 threads = 8 wave32s per WGP
// slot; 128 blocks -> 1024 waves in flight to hide random-gather latency.
__global__ __launch_bounds__(256) void tl_partial(
    const float* __restrict__ cls,      // (BS, 2, N_A) after reshape semantics
    const float* __restrict__ loc,      // (BS, 4, N_A)
    const float* __restrict__ gts,      // (BS, 4)
    const int*   __restrict__ pos_idx,  // (BS, POS)
    const int*   __restrict__ neg_idx,  // (BS, NEG)
    const float* __restrict__ anchors,  // (N_A, 4) [cx, cy, w, h]
    float*       __restrict__ part)     // (3, BS) per-batch means
{
    const int b   = blockIdx.x;
    const int tid = threadIdx.x;
    const float* clsb = cls + (size_t)b * 2u * N_A;
    const float* locb = loc + (size_t)b * 4u * N_A;

    float neg_acc = 0.f, pos_acc = 0.f, loc_acc = 0.f;

    // Negative-anchor classification loss: -log_softmax(...)[0]
    for (int j = tid; j < NEG; j += 256) {
        const int a = neg_idx[b * NEG + j];
        const float l0 = clsb[a];
        const float l1 = clsb[N_A + a];
        neg_acc += softplus_f(l1 - l0);
    }

    // Positive anchors: classification loss (-log p1) + smooth-L1 loc loss
    if (tid < POS) {
        const int a = pos_idx[b * POS + tid];
        const float l0 = clsb[a];
        const float l1 = clsb[N_A + a];
        pos_acc = softplus_f(l0 - l1);

        const float4 an = ((const float4*)anchors)[a];   // cx, cy, w, h
        const float gx = gts[b * 4 + 0];
        const float gy = gts[b * 4 + 1];
        const float gw = gts[b * 4 + 2];
        const float gh = gts[b * 4 + 3];
        const float t0 = (gx - an.x) / an.z;
        const float t1 = (gy - an.y) / an.w;
        const float t2 = logf(gw / an.z);
        const float t3 = logf(gh / an.w);

        const float p0 = locb[0 * N_A + a];
        const float p1 = locb[1 * N_A + a];
        const float p2 = locb[2 * N_A + a];
        const float p3 = locb[3 * N_A + a];

        float s = 0.f;
        #define SL1(p, t) { float d_ = (p) - (t); float ad_ = fabsf(d_); \
                            s += (ad_ < 1.f) ? 0.5f * d_ * d_ : ad_ - 0.5f; }
        SL1(p0, t0) SL1(p1, t1) SL1(p2, t2) SL1(p3, t3)
        #undef SL1
        loc_acc = s;
    }

    // wave32 reduction (gfx1250: warpSize == 32)
    #pragma unroll
    for (int off = 16; off >= 1; off >>= 1) {
        neg_acc += __shfl_down(neg_acc, off, 32);
        pos_acc += __shfl_down(pos_acc, off, 32);
        loc_acc += __shfl_down(loc_acc, off, 32);
    }

    __shared__ float red[8][3];
    const int wave = tid >> 5;
    if ((tid & 31) == 0) {
        red[wave][0] = neg_acc;
        red[wave][1] = pos_acc;
        red[wave][2] = loc_acc;
    }
    __syncthreads();
    if (tid == 0) {
        float n = 0.f, p = 0.f, l = 0.f;
        #pragma unroll
        for (int w = 0; w < 8; ++w) { n += red[w][0]; p += red[w][1]; l += red[w][2]; }
        part[0 * BS + b] = n * (1.0f / NEG);
        part[1 * BS + b] = p * (1.0f / POS);
        part[2 * BS + b] = l * (1.0f / (POS * 4));
    }
}

// Kernel 2: exact 128-way batch reduction per component using the matrix
// engine. With A = all-ones (16x4), D[m][n] = sum_k B[k][n], so summing D row 0
// over its 16 lanes yields the sum of ALL 64 B entries regardless of the exact
// (k,n)<->VGPR mapping. Two C-chained WMMAs reduce 128 values in full f32.
// One full wave32, no divergence around the WMMAs (EXEC must be all ones).
__global__ __launch_bounds__(32) void tl_final(
    const float* __restrict__ part, float* __restrict__ out)
{
    const int lane = threadIdx.x;   // 0..31

    v2f ones; ones.x = 1.0f; ones.y = 1.0f;
    float sums[3];

    #pragma unroll
    for (int c = 0; c < 3; ++c) {
        const float* x = part + c * BS;
        v2f b0, b1;
        b0.x = x[lane];        b0.y = x[lane + 32];
        b1.x = x[lane + 64];   b1.y = x[lane + 96];

        v8f acc = {};
        // D = ones(16x4) * B(4x16) + C  -> column sums, accumulated over both halves
        acc = __builtin_amdgcn_wmma_f32_16x16x4_f32(
            false, ones, false, b0, (short)0, acc, false, false);
        acc = __builtin_amdgcn_wmma_f32_16x16x4_f32(
            false, ones, false, b1, (short)0, acc, false, false);

        // acc[0]: lanes 0-15 hold D[0][n], lanes 16-31 hold D[8][n] (identical
        // rows since A's rows are identical). Sum the 16 columns per half-wave.
        float r = acc[0];
        r += __shfl_xor(r, 1, 32);
        r += __shfl_xor(r, 2, 32);
        r += __shfl_xor(r, 4, 32);
        r += __shfl_xor(r, 8, 32);
        sums[c] = r;
    }

    if (lane == 0) {
        const float neg_m = sums[0] * (1.0f / BS);
        const float pos_m = sums[1] * (1.0f / BS);
        const float loc_m = sums[2] * (1.0f / BS);
        const float loss  = 0.5f * (0.5f * neg_m + 0.5f * pos_m) + 0.5f * loc_m;
        out[0] = loss;
        out[1] = neg_m;
        out[2] = pos_m;
        out[3] = loc_m;
    }
}

extern "C" void kernel_launch(void* const* d_in, const int* in_sizes, int n_in,
                              void* d_out, int out_size, void* d_ws, size_t ws_size,
                              hipStream_t stream) {
    const float* cls  = (const float*)d_in[0];  // (128, 32, 64, 64)
    const float* loc  = (const float*)d_in[1];  // (128, 64, 64, 64)
    const float* gts  = (const float*)d_in[2];  // (128, 4)
    const int*   pos  = (const int*)  d_in[3];  // (128, 64)
    const int*   neg  = (const int*)  d_in[4];  // (128, 2048)
    const float* anch = (const float*)d_in[5];  // (65536, 4)

    float* part = (float*)d_ws;                 // 3 * 128 floats, rewritten each call

    tl_partial<<<BS, 256, 0, stream>>>(cls, loc, gts, pos, neg, anch, part);
    tl_final<<<1, 32, 0, stream>>>(part, (float*)d_out);
}